// ImprovedGCNEncoder_13520557048097
// MI455X (gfx1250) — compile-verified
//
#include <hip/hip_runtime.h>

typedef float v2f __attribute__((ext_vector_type(2)));
typedef float v8f __attribute__((ext_vector_type(8)));

#define N_NODES 50000
#define N_EDGES 800000

// ---------------------------------------------------------------------------
// Degree / normalization
// ---------------------------------------------------------------------------
__global__ void deg_init(float* __restrict__ dinv, int n) {
    int i = blockIdx.x * blockDim.x + threadIdx.x;
    if (i < n) dinv[i] = 1.0f;  // self-loop contributes 1 to every node's degree
}

__global__ void deg_count(float* __restrict__ dinv, const long long* __restrict__ ei, int e) {
    int i = blockIdx.x * blockDim.x + threadIdx.x;
    if (i < e) {
        int dst = (int)ei[N_EDGES + i];
        atomicAdd(&dinv[dst], 1.0f);
    }
}

__global__ void deg_rsqrt(float* __restrict__ dinv, int n) {
    int i = blockIdx.x * blockDim.x + threadIdx.x;
    if (i < n) dinv[i] = rsqrtf(dinv[i]);  // deg >= 1 always (self loops)
}

// ---------------------------------------------------------------------------
// Dense transform: H[M,Nout] = A[M,K] @ W[K,Nout], all f32 row-major.
// Full-precision f32 WMMA (V_WMMA_F32_16X16X4_F32), wave32.
// M % 16 == 0, K % 8 == 0, Nout % 16 == 0 guaranteed by problem sizes.
// Block = 128 threads (4 waves); block owns a 16-row panel of A in LDS.
// ---------------------------------------------------------------------------
__global__ __launch_bounds__(128)
void gemm_f32_wmma(const float* __restrict__ A, const float* __restrict__ W,
                   float* __restrict__ H, int K, int Nout) {
    __shared__ float Alds[16 * 260];           // max K=256, pitch K+4 kills bank conflicts
    const int pitch = K + 4;
    const int m0  = blockIdx.x << 4;
    const int tid = threadIdx.x;

    // Coalesced stage of the 16 x K A-panel
    for (int idx = tid; idx < 16 * K; idx += 128) {
        int r = idx / K;
        int c = idx - r * K;
        Alds[r * pitch + c] = A[(size_t)(m0 + r) * K + c];
    }
    __syncthreads();

    const int wave = tid >> 5;
    const int lane = tid & 31;
    const int half = lane >> 4;   // 0: lanes 0-15 (K+0,K+1), 1: lanes 16-31 (K+2,K+3)
    const int lm   = lane & 15;
    const int ntiles = Nout >> 4;

    for (int t = wave; t < ntiles; t += 4) {
        const int n = (t << 4) + lm;
        v8f c0 = {};
        v8f c1 = {};
        for (int k = 0; k < K; k += 8) {
            // A 16x4 layout (ISA 7.12.2): lane half selects K pair, row = lane&15
            v2f a0 = *(const v2f*)&Alds[lm * pitch + k + 2 * half];
            v2f b0;
            b0.x = W[(size_t)(k + 2 * half + 0) * Nout + n];
            b0.y = W[(size_t)(k + 2 * half + 1) * Nout + n];
            c0 = __builtin_amdgcn_wmma_f32_16x16x4_f32(false, a0, false, b0,
                                                       (short)0, c0, false, false);
            v2f a1 = *(const v2f*)&Alds[lm * pitch + k + 4 + 2 * half];
            v2f b1;
            b1.x = W[(size_t)(k + 4 + 2 * half + 0) * Nout + n];
            b1.y = W[(size_t)(k + 4 + 2 * half + 1) * Nout + n];
            c1 = __builtin_amdgcn_wmma_f32_16x16x4_f32(false, a1, false, b1,
                                                       (short)0, c1, false, false);
        }
        v8f c = c0 + c1;
#pragma unroll
        for (int r = 0; r < 8; ++r) {
            // C/D layout: VGPR r -> row r (lanes 0-15) / row r+8 (lanes 16-31)
            H[(size_t)(m0 + r + 8 * half) * Nout + n] = c[r];
        }
    }
}

// ---------------------------------------------------------------------------
// Aggregation: agg = selfloop + sum_edges, then relu(agg + bias)
// ---------------------------------------------------------------------------
__global__ void agg_self(const float* __restrict__ h, const float* __restrict__ dinv,
                         float* __restrict__ agg, int total, int F) {
    int i = blockIdx.x * blockDim.x + threadIdx.x;
    if (i >= total) return;
    int node = i / F;
    float d = dinv[node];
    agg[i] = h[i] * d * d;  // self-loop edge: norm = dinv[i]^2
}

// lshift = log2(F/4): group of F/4 lanes per edge; each lane moves a float4
__global__ void agg_edges(const float* __restrict__ h, float* __restrict__ agg,
                          const float* __restrict__ dinv, const long long* __restrict__ ei,
                          int F, int lshift) {
    int gid  = blockIdx.x * blockDim.x + threadIdx.x;
    int e    = gid >> lshift;
    int lane = gid & ((1 << lshift) - 1);
    if (e >= N_EDGES) return;
    int src = (int)ei[e];
    int dst = (int)ei[N_EDGES + e];
    float nrm = dinv[src] * dinv[dst];
    int base = lane << 2;
    const float4 hv = *(const float4*)(h + (size_t)src * F + base);
    float* ap = agg + (size_t)dst * F + base;
    atomicAdd(ap + 0, hv.x * nrm);
    atomicAdd(ap + 1, hv.y * nrm);
    atomicAdd(ap + 2, hv.z * nrm);
    atomicAdd(ap + 3, hv.w * nrm);
}

__global__ void finalize(const float* __restrict__ agg, const float* __restrict__ bias,
                         float* __restrict__ out, int total, int F) {
    int i = blockIdx.x * blockDim.x + threadIdx.x;
    if (i >= total) return;
    int col = i % F;
    out[i] = fmaxf(agg[i] + bias[col], 0.0f);
}

// ---------------------------------------------------------------------------
// Host-side orchestration
// ---------------------------------------------------------------------------
static inline void run_layer(const float* in, const float* W, const float* b,
                             float* h, float* agg, float* out,
                             const float* dinv, const long long* ei,
                             int K, int F, hipStream_t stream) {
    // 1) h = in @ W  (WMMA f32)
    gemm_f32_wmma<<<N_NODES / 16, 128, 0, stream>>>(in, W, h, K, F);
    // 2) agg = selfloop
    int total = N_NODES * F;
    agg_self<<<(total + 255) / 256, 256, 0, stream>>>(h, dinv, agg, total, F);
    // 3) scatter-add over edges
    int lshift = (F == 128) ? 5 : 4;           // lanes per edge = F/4
    long long threads = (long long)N_EDGES << lshift;
    agg_edges<<<(unsigned)((threads + 255) / 256), 256, 0, stream>>>(h, agg, dinv, ei, F, lshift);
    // 4) out = relu(agg + b)
    finalize<<<(total + 255) / 256, 256, 0, stream>>>(agg, b, out, total, F);
}

extern "C" void kernel_launch(void* const* d_in, const int* in_sizes, int n_in,
                              void* d_out, int out_size, void* d_ws, size_t ws_size,
                              hipStream_t stream) {
    const float*     x  = (const float*)d_in[0];
    const long long* ei = (const long long*)d_in[1];   // int64 edge_index [2, E]
    const float*     W1 = (const float*)d_in[2];
    const float*     b1 = (const float*)d_in[3];
    const float*     W2 = (const float*)d_in[4];
    const float*     b2 = (const float*)d_in[5];
    const float*     W3 = (const float*)d_in[6];
    const float*     b3 = (const float*)d_in[7];
    float* out = (float*)d_out;

    float* ws   = (float*)d_ws;
    float* dinv = ws;                         // 50000
    float* h    = ws + 50048;                 // up to 50000*128
    float* agg  = h + (size_t)N_NODES * 128;  // up to 50000*128
    float* xbuf = agg + (size_t)N_NODES * 128;

    // Normalization coefficients (shared by all 3 layers)
    deg_init <<<(N_NODES + 255) / 256, 256, 0, stream>>>(dinv, N_NODES);
    deg_count<<<(N_EDGES + 255) / 256, 256, 0, stream>>>(dinv, ei, N_EDGES);
    deg_rsqrt<<<(N_NODES + 255) / 256, 256, 0, stream>>>(dinv, N_NODES);

    // Layer 1: 256 -> 128
    run_layer(x,    W1, b1, h, agg, xbuf, dinv, ei, 256, 128, stream);
    // Layer 2: 128 -> 128 (xbuf read in GEMM before being overwritten by finalize)
    run_layer(xbuf, W2, b2, h, agg, xbuf, dinv, ei, 128, 128, stream);
    // Layer 3: 128 -> 64, write final output
    run_layer(xbuf, W3, b3, h, agg, out,  dinv, ei, 128, 64, stream);
}